// BEVLoss_18846316495052
// MI455X (gfx1250) — compile-verified
//
#include <hip/hip_runtime.h>
#include <hip/hip_bf16.h>

// ---------------------------------------------------------------------------
// BEVLoss for MI455X (gfx1250).
// Bandwidth-bound streaming reductions; WMMA f32 16x16x4 used as the
// cross-lane wave-reduction engine (ones-matrix trick), keeping fp32 precision
// and exercising the CDNA5 matrix pipe. Feat kernel streams with NT hints.
// ---------------------------------------------------------------------------

typedef __attribute__((ext_vector_type(2))) float v2f;
typedef __attribute__((ext_vector_type(4))) float v4f;
typedef __attribute__((ext_vector_type(8))) float v8f;

// Full 32-lane sum of (a0 + a1) using V_WMMA_F32_16X16X4_F32.
// Stage 1: A = lane partials (A[m][0]=a0[lane m], A[m][1]=a1[lane m],
//          A[m][2]=a0[lane m+16], A[m][3]=a1[lane m+16]), B = ones
//          -> D1 row m = partial sum of lanes {m, m+16}.
// Stage 2: A = ones, B = successive VGPR pairs of D1 (4 accumulating WMMAs);
//          the 8 VGPRs of D1 cover all 16 rows, so every element of D2 equals
//          the full 32-lane total. EXEC must be all 1s at the call site.
__device__ __forceinline__ float wave_sum_wmma(float a0, float a1) {
    v2f A;    A[0] = a0;   A[1] = a1;
    v2f ones; ones[0] = 1.0f; ones[1] = 1.0f;
    v8f z = {};
    v8f d1 = __builtin_amdgcn_wmma_f32_16x16x4_f32(false, A, false, ones,
                                                   (short)0, z, false, false);
    v8f d2 = {};
    v2f b;
    b[0] = d1[0]; b[1] = d1[1];
    d2 = __builtin_amdgcn_wmma_f32_16x16x4_f32(false, ones, false, b,
                                               (short)0, d2, false, false);
    b[0] = d1[2]; b[1] = d1[3];
    d2 = __builtin_amdgcn_wmma_f32_16x16x4_f32(false, ones, false, b,
                                               (short)0, d2, false, false);
    b[0] = d1[4]; b[1] = d1[5];
    d2 = __builtin_amdgcn_wmma_f32_16x16x4_f32(false, ones, false, b,
                                               (short)0, d2, false, false);
    b[0] = d1[6]; b[1] = d1[7];
    d2 = __builtin_amdgcn_wmma_f32_16x16x4_f32(false, ones, false, b,
                                               (short)0, d2, false, false);
    return d2[0];
}

// Block-wide sum broadcast to all threads (blockDim.x must be a multiple of 32,
// <= 256). Deterministic: fixed summation order.
__device__ __forceinline__ float block_sum_all(float v) {
    __shared__ float smem[9];
    const int wave = threadIdx.x >> 5;
    const int lane = threadIdx.x & 31;
    float w = wave_sum_wmma(v, 0.0f);
    __syncthreads();                 // protect smem reuse across invocations
    if (lane == 0) smem[wave] = w;
    __syncthreads();
    if (threadIdx.x == 0) {
        float s = 0.0f;
        const int nw = (int)(blockDim.x >> 5);
        for (int i = 0; i < nw; ++i) s += smem[i];
        smem[8] = s;
    }
    __syncthreads();
    return smem[8];
}

__device__ __forceinline__ void block_reduce_store(float a0, float a1,
                                                   float* __restrict__ out,
                                                   int idx) {
    float s = block_sum_all(a0 + a1);
    if (threadIdx.x == 0) out[idx] = s;
}

__device__ __forceinline__ float bce_term(float x, float t) {
    return fmaxf(x, 0.0f) - x * t + log1pf(expf(-fabsf(x)));
}

// ---------------------------------------------------------------------------
// Kernel 1: feat_loss partials. Streams bev & pos once with non-temporal
// b128 loads (TH=NT): 655 MB single-use traffic should not pollute WGP$/L2.
// 8 floats (32 B) per lane per iteration -> 4 independent b128 loads in
// flight per iteration for HBM latency hiding. 81.92M floats = 10.24M chunks
// exactly (no tail).
// ---------------------------------------------------------------------------
__global__ void feat_partial_kernel(const float* __restrict__ bev,
                                    const float* __restrict__ pos,
                                    float* __restrict__ ws, long nchunk) {
    float a0 = 0.0f, a1 = 0.0f;
    const long stride = (long)gridDim.x * (long)blockDim.x;
    for (long c = (long)blockIdx.x * blockDim.x + threadIdx.x; c < nchunk;
         c += stride) {
        const v4f* bp = (const v4f*)(bev + (c << 3));
        const v4f* pp = (const v4f*)(pos + (c << 3));
        v4f x0 = __builtin_nontemporal_load(bp);
        v4f x1 = __builtin_nontemporal_load(bp + 1);
        v4f p0 = __builtin_nontemporal_load(pp);
        v4f p1 = __builtin_nontemporal_load(pp + 1);
        v4f d0 = x0 - p0;
        v4f d1 = x1 - p1;
        a0 += d0.x * d0.x + d0.y * d0.y + d0.z * d0.z + d0.w * d0.w;
        a1 += d1.x * d1.x + d1.y * d1.y + d1.z * d1.z + d1.w * d1.w;
    }
    block_reduce_store(a0, a1, ws, blockIdx.x);
}

// ---------------------------------------------------------------------------
// Kernel 2: lane_loss partials. Fuses 400x400 -> 200x200 bilinear target.
// Grid: exactly 320000 threads (1250 blocks x 256).
// ---------------------------------------------------------------------------
__global__ void lane_partial_kernel(const float* __restrict__ bev,
                                    const float* __restrict__ masks,
                                    float* __restrict__ ws) {
    const int p   = blockIdx.x * blockDim.x + threadIdx.x;   // 0..319999
    const int b   = p / 40000;
    const int rem = p % 40000;
    const int oy  = rem / 200;
    const int ox  = rem % 200;

    const float scale = 399.0f / 199.0f;     // linspace(0, 399, 200) step
    const float ysf = (float)oy * scale;
    const float xsf = (float)ox * scale;
    const int y0 = (int)floorf(ysf);
    const int x0 = (int)floorf(xsf);
    const int y1 = min(y0 + 1, 399);
    const int x1 = min(x0 + 1, 399);
    const float wy = ysf - (float)y0;
    const float wx = xsf - (float)x0;

    const float* m = masks + (size_t)b * 160000;
    const float v00 = m[y0 * 400 + x0];
    const float v01 = m[y0 * 400 + x1];
    const float v10 = m[y1 * 400 + x0];
    const float v11 = m[y1 * 400 + x1];
    const float t = v00 * (1.0f - wy) * (1.0f - wx)
                  + v01 * (1.0f - wy) * wx
                  + v10 * wy * (1.0f - wx)
                  + v11 * wy * wx;

    const float l = bev[(size_t)b * 256 * 40000 + rem];      // channel 0
    block_reduce_store(bce_term(l, t), 0.0f, ws, blockIdx.x);
}

// ---------------------------------------------------------------------------
// Kernel 3: obj_loss partials. Fuses gaussian box-heatmap (max over boxes).
// Grid: (157, 8) blocks x 256 threads; blockIdx.y = batch.
// ---------------------------------------------------------------------------
__global__ void obj_partial_kernel(const float* __restrict__ bev,
                                   const float* __restrict__ boxes,  // (8,64,4)
                                   const int* __restrict__ valid,    // (8,)
                                   float* __restrict__ ws, int gridX) {
    const int b = blockIdx.y;
    __shared__ float fx[64], fy[64], fs[64];
    __shared__ int   xlo[64], xhi[64], ylo[64], yhi[64];
    __shared__ int   nv_s;

    if (threadIdx.x == 0) nv_s = min(valid[b], 64);
    if (threadIdx.x < 64) {
        const float* bx = boxes + ((size_t)b * 64 + threadIdx.x) * 4;
        const int x = (int)floorf(bx[0] * 200.0f / 600.0f);
        const int y = (int)floorf(bx[1] * 200.0f / 600.0f);
        const int w = (int)floorf(bx[2] * 200.0f / 600.0f);
        const int h = (int)floorf(bx[3] * 200.0f / 600.0f);
        const float sigma = (float)min(w, h) / 6.0f;
        fx[threadIdx.x] = (float)x;
        fy[threadIdx.x] = (float)y;
        fs[threadIdx.x] = 1.0f / sigma;
        xlo[threadIdx.x] = max(0, x - w / 2);
        xhi[threadIdx.x] = min(200, x + w / 2);
        ylo[threadIdx.x] = max(0, y - h / 2);
        yhi[threadIdx.x] = min(200, y + h / 2);
    }
    __syncthreads();

    const int pix = blockIdx.x * blockDim.x + threadIdx.x;
    float term = 0.0f;
    if (pix < 40000) {
        const int j = pix / 200;   // row (J over H)
        const int i = pix % 200;   // col (I over W)
        float hm = 0.0f;
        const int nv = nv_s;
        for (int n = 0; n < nv; ++n) {
            if (i >= xlo[n] && i < xhi[n] && j >= ylo[n] && j < yhi[n]) {
                const float dx = ((float)i - fx[n]) * fs[n];
                const float dy = ((float)j - fy[n]) * fs[n];
                const float g = expf(-0.5f * (dx * dx + dy * dy));
                hm = fmaxf(hm, g);
            }
        }
        const float l = bev[((size_t)b * 256 + 1) * 40000 + pix];  // channel 1
        term = bce_term(l, hm);
    }
    // All 256 threads reach here (tail threads contribute 0) -> EXEC all ones.
    block_reduce_store(term, 0.0f, ws, blockIdx.y * gridX + blockIdx.x);
}

// ---------------------------------------------------------------------------
// Kernel 4: finalize. Single block; deterministic summation of partials.
// Output order matches the reference tuple: (total, lane, obj, feat).
// ---------------------------------------------------------------------------
__global__ void finalize_kernel(const float* __restrict__ wsf, int nf,
                                const float* __restrict__ wsl, int nl,
                                const float* __restrict__ wso, int no,
                                float* __restrict__ out) {
    float sf = 0.0f, sl = 0.0f, so = 0.0f;
    for (int i = threadIdx.x; i < nf; i += blockDim.x) sf += wsf[i];
    for (int i = threadIdx.x; i < nl; i += blockDim.x) sl += wsl[i];
    for (int i = threadIdx.x; i < no; i += blockDim.x) so += wso[i];
    const float Sf = block_sum_all(sf);
    const float Sl = block_sum_all(sl);
    const float So = block_sum_all(so);
    if (threadIdx.x == 0) {
        const float lane = Sl / 320000.0f;
        const float obj  = So / 320000.0f;
        const float feat = Sf / 81920000.0f;
        out[0] = lane + obj + 0.1f * feat;
        out[1] = lane;
        out[2] = obj;
        out[3] = feat;
    }
}

// ---------------------------------------------------------------------------
extern "C" void kernel_launch(void* const* d_in, const int* in_sizes, int n_in,
                              void* d_out, int out_size, void* d_ws, size_t ws_size,
                              hipStream_t stream) {
    const float* bev   = (const float*)d_in[0];   // (8,256,200,200) f32
    const float* pos   = (const float*)d_in[1];   // (8,256,200,200) f32
    const float* masks = (const float*)d_in[2];   // (8,400,400) f32
    const float* boxes = (const float*)d_in[3];   // (8,64,4) f32
    const int*   valid = (const int*)d_in[4];     // (8,) i32
    float* out = (float*)d_out;
    float* ws  = (float*)d_ws;

    const int NB1 = 4096;           // feat partial blocks
    const int NB2 = 1250;           // lane partial blocks (320000 / 256)
    const int GX3 = 157;            // ceil(40000 / 256)
    const int NB3 = GX3 * 8;

    float* wsf = ws;
    float* wsl = ws + NB1;
    float* wso = ws + NB1 + NB2;

    const long nchunk = 81920000L / 8;  // 8-float (32 B) chunks, exact

    feat_partial_kernel<<<NB1, 256, 0, stream>>>(bev, pos, wsf, nchunk);
    lane_partial_kernel<<<NB2, 256, 0, stream>>>(bev, masks, wsl);
    obj_partial_kernel<<<dim3(GX3, 8), 256, 0, stream>>>(bev, boxes, valid, wso, GX3);
    finalize_kernel<<<1, 256, 0, stream>>>(wsf, NB1, wsl, NB2, wso, NB3, out);
}